// GatedGCNLayer_8821862826043
// MI455X (gfx1250) — compile-verified
//
#include <hip/hip_runtime.h>

// GatedGCN layer for MI455X (gfx1250), fp32 throughout (matches reference dtype).
// GEMMs via V_WMMA_F32_16X16X4_F32; edge GEMM fused with message+sigmoid+scatter.
// W is staged in LDS k-pair-interleaved so each B fragment is one ds_load_b64
// straight into an even-aligned VGPR pair (no repacking movs).

#define N_NODES 50000
#define N_EDGES 500000
#define D 128
#define BN_EPS 1e-5f

typedef float v2f __attribute__((ext_vector_type(2)));
typedef float v8f __attribute__((ext_vector_type(8)));

__device__ __forceinline__ v8f wmma_f32(v2f a, v2f b, v8f c) {
  // 8 args: (neg_a, A, neg_b, B, c_mod, C, reuse_a, reuse_b)
  return __builtin_amdgcn_wmma_f32_16x16x4_f32(false, a, false, b, (short)0, c,
                                               false, false);
}

// Stage W[128][128] into LDS as k-pair interleaved float2:
//   sW2[p*128 + n] = { W[2p][n], W[2p+1][n] },  p in [0,64)
__device__ __forceinline__ void stage_w_interleaved(const float* __restrict__ W,
                                                    float* __restrict__ sW,
                                                    int tid) {
  float2* sW2 = (float2*)sW;
  for (int i = tid; i < 64 * D; i += 256) {
    const int p = i >> 7, n = i & (D - 1);
    float2 w;
    w.x = W[(2 * p) * D + n];
    w.y = W[(2 * p + 1) * D + n];
    sW2[i] = w;
  }
}

// ---------------------------------------------------------------------------
// C[M,128] = A[M,128] @ W[128,128] + bias.  8 waves/block, 16 rows per wave.
// ---------------------------------------------------------------------------
__global__ __launch_bounds__(256) void gemm_node_kernel(
    const float* __restrict__ A, const float* __restrict__ W,
    const float* __restrict__ bias, float* __restrict__ C, int M) {
  __shared__ float sW[D * D];  // 64 KB, k-pair interleaved
  const int tid = threadIdx.x;
  stage_w_interleaved(W, sW, tid);
  __syncthreads();

  const int wave = tid >> 5, lane = tid & 31;
  const int l15 = lane & 15, hi = lane >> 4;
  const int row0 = (blockIdx.x * 8 + wave) * 16;
  if (row0 < M) {
    v8f acc[8];
#pragma unroll
    for (int t = 0; t < 8; ++t) {
      const float bv = bias[t * 16 + l15];
#pragma unroll
      for (int r = 0; r < 8; ++r) acc[t][r] = bv;
    }

    const float* Arow = A + (long)(row0 + l15) * D + 2 * hi;
    const v2f* sWB = (const v2f*)sW;       // [64][128] float2
    const int bb = hi * D + l15;           // lane base (float2 units)
#pragma unroll 4
    for (int kk = 0; kk < D; kk += 4) {
      const v2f a = *(const v2f*)(Arow + kk);  // A frag: K = kk+2*hi .. +1
      const int bk = (kk >> 1) * D + bb;       // (kk/2 + hi)*128 + l15
#pragma unroll
      for (int t = 0; t < 8; ++t) {
        const v2f b = sWB[bk + t * 16];        // one ds_load_b64
        acc[t] = wmma_f32(a, b, acc[t]);
      }
    }

#pragma unroll
    for (int r = 0; r < 8; ++r) {
      float* Crow = C + (long)(row0 + r + 8 * hi) * D;
#pragma unroll
      for (int t = 0; t < 8; ++t) Crow[t * 16 + l15] = acc[t][r];
    }
  }
}

// ---------------------------------------------------------------------------
// Fused edge pipeline:  e = E_X@WC + bC + DX[src] + EX[dst];  sig = sigmoid(e)
//   num[dst] += sig*BX[src];  den[dst] += sig;  Epre = e*snorm_e (to out_E)
//   + per-column sum / sumsq of Epre for the E BatchNorm.
// Never materializes CE (saves 512 MB of HBM traffic).
// ---------------------------------------------------------------------------
__global__ __launch_bounds__(256) void gemm_edge_kernel(
    const float* __restrict__ EXin, const float* __restrict__ W,
    const float* __restrict__ bias, const float* __restrict__ DXm,
    const float* __restrict__ EXm, const float* __restrict__ BXm,
    const int* __restrict__ src, const int* __restrict__ dst,
    const float* __restrict__ snorm_e, float* __restrict__ num,
    float* __restrict__ den, float* __restrict__ Epre,
    float* __restrict__ sums /* [0..127]=sum, [128..255]=sumsq */) {
  __shared__ float sW[D * D];  // 64 KB, reused for BN partials after k-loop
  const int tid = threadIdx.x;
  stage_w_interleaved(W, sW, tid);
  __syncthreads();

  const int wave = tid >> 5, lane = tid & 31;
  const int l15 = lane & 15, hi = lane >> 4;
  const int row0 = (blockIdx.x * 8 + wave) * 16;
  const bool active = (row0 < N_EDGES);

  v8f acc[8];
  if (active) {
#pragma unroll
    for (int t = 0; t < 8; ++t) {
      const float bv = bias[t * 16 + l15];
#pragma unroll
      for (int r = 0; r < 8; ++r) acc[t][r] = bv;
    }
    const float* Arow = EXin + (long)(row0 + l15) * D + 2 * hi;
    const v2f* sWB = (const v2f*)sW;
    const int bb = hi * D + l15;
#pragma unroll 4
    for (int kk = 0; kk < D; kk += 4) {
      const v2f a = *(const v2f*)(Arow + kk);
      const int bk = (kk >> 1) * D + bb;
#pragma unroll
      for (int t = 0; t < 8; ++t) {
        const v2f b = sWB[bk + t * 16];
        acc[t] = wmma_f32(a, b, acc[t]);
      }
    }
  }
  __syncthreads();  // all waves done reading sW -> reuse as stats scratch
  float* sSum = sW;
  float* sSq = sW + D;
  if (tid < 2 * D) sW[tid] = 0.0f;
  __syncthreads();

  if (active) {
    float lsum[8], lsq[8];
#pragma unroll
    for (int t = 0; t < 8; ++t) { lsum[t] = 0.0f; lsq[t] = 0.0f; }
#pragma unroll
    for (int r = 0; r < 8; ++r) {
      const int row = row0 + r + 8 * hi;
      const int s = src[row];
      const int d2 = dst[row];
      const float sn = snorm_e[row];
      const float* DXr = DXm + (long)s * D;
      const float* EXr = EXm + (long)d2 * D;
      const float* BXr = BXm + (long)s * D;
      float* numr = num + (long)d2 * D;
      float* denr = den + (long)d2 * D;
      float* Er = Epre + (long)row * D;
#pragma unroll
      for (int t = 0; t < 8; ++t) {
        const int col = t * 16 + l15;
        const float e = acc[t][r] + DXr[col] + EXr[col];
        const float sg = 1.0f / (1.0f + __expf(-e));
        atomicAdd(&numr[col], sg * BXr[col]);  // global_atomic_add_f32 -> L2
        atomicAdd(&denr[col], sg);
        const float ep = e * sn;
        Er[col] = ep;
        lsum[t] += ep;
        lsq[t] += ep * ep;
      }
    }
#pragma unroll
    for (int t = 0; t < 8; ++t) {
      atomicAdd(&sSum[t * 16 + l15], lsum[t]);  // ds_add_f32
      atomicAdd(&sSq[t * 16 + l15], lsq[t]);
    }
  }
  __syncthreads();
  if (tid < D) {
    atomicAdd(&sums[tid], sSum[tid]);
    atomicAdd(&sums[D + tid], sSq[tid]);
  }
}

// ---------------------------------------------------------------------------
// h = (den>0) ? AX + num/den : X   (sigmoid>0 => den>0 iff in-degree>0)
// Hpre = h * snorm_n  (to out_H region) + per-column BN sums for H.
// ---------------------------------------------------------------------------
__global__ __launch_bounds__(256) void node_combine_kernel(
    const float* __restrict__ X, const float* __restrict__ AX,
    const float* __restrict__ num, const float* __restrict__ den,
    const float* __restrict__ snorm_n, float* __restrict__ Hpre,
    float* __restrict__ sums) {
  __shared__ float sSum[D], sSq[D];
  const int tid = threadIdx.x;
  if (tid < D) { sSum[tid] = 0.0f; sSq[tid] = 0.0f; }
  __syncthreads();

  const int total = N_NODES * D;
  const int stride = gridDim.x * 256;  // multiple of 128 -> col invariant
  const int i0 = blockIdx.x * 256 + tid;
  const int col = i0 & (D - 1);
  float ls = 0.0f, lq = 0.0f;
  for (int i = i0; i < total; i += stride) {
    const int row = i >> 7;
    const float dv = den[i];
    const float h = (dv > 0.0f) ? (AX[i] + num[i] / dv) : X[i];
    const float hp = h * snorm_n[row];
    Hpre[i] = hp;
    ls += hp;
    lq += hp * hp;
  }
  atomicAdd(&sSum[col], ls);
  atomicAdd(&sSq[col], lq);
  __syncthreads();
  if (tid < D) {
    atomicAdd(&sums[tid], sSum[tid]);
    atomicAdd(&sums[D + tid], sSq[tid]);
  }
}

// ---------------------------------------------------------------------------
// Fold BN stats into per-column affine: a = gamma*rsqrt(var+eps), b = beta-m*a
// ---------------------------------------------------------------------------
__global__ void bn_params_kernel(const float* __restrict__ stats,
                                 const float* __restrict__ gamma_h,
                                 const float* __restrict__ beta_h,
                                 const float* __restrict__ gamma_e,
                                 const float* __restrict__ beta_e,
                                 float* __restrict__ ab) {
  const int tid = threadIdx.x;  // 256 threads
  if (tid < D) {
    const float m = stats[tid] * (1.0f / (float)N_NODES);
    const float v = stats[D + tid] * (1.0f / (float)N_NODES) - m * m;
    const float a = gamma_h[tid] * rsqrtf(v + BN_EPS);
    ab[tid] = a;
    ab[D + tid] = beta_h[tid] - m * a;
  } else if (tid < 2 * D) {
    const int c = tid - D;
    const float m = stats[2 * D + c] * (1.0f / (float)N_EDGES);
    const float v = stats[3 * D + c] * (1.0f / (float)N_EDGES) - m * m;
    const float a = gamma_e[c] * rsqrtf(v + BN_EPS);
    ab[2 * D + c] = a;
    ab[3 * D + c] = beta_e[c] - m * a;
  }
}

// ---------------------------------------------------------------------------
// out = base + relu(pre*a[col] + b[col]); pre lives in `out` (in-place).
// ---------------------------------------------------------------------------
__global__ __launch_bounds__(256) void finalize_kernel(
    const float* __restrict__ base, float* __restrict__ out,
    const float* __restrict__ a, const float* __restrict__ b, int total4) {
  const float4* base4 = (const float4*)base;
  float4* out4 = (float4*)out;
  const float4* a4 = (const float4*)a;
  const float4* b4 = (const float4*)b;
  const int stride = gridDim.x * 256;  // multiple of 32 -> col4 invariant
  const int i0 = blockIdx.x * 256 + threadIdx.x;
  const int c4 = i0 & 31;
  const float4 av = a4[c4], bv = b4[c4];
  for (int i = i0; i < total4; i += stride) {
    const float4 p = out4[i];
    const float4 x = base4[i];
    float4 o;
    o.x = x.x + fmaxf(0.0f, p.x * av.x + bv.x);
    o.y = x.y + fmaxf(0.0f, p.y * av.y + bv.y);
    o.z = x.z + fmaxf(0.0f, p.z * av.z + bv.z);
    o.w = x.w + fmaxf(0.0f, p.w * av.w + bv.w);
    out4[i] = o;
  }
}

extern "C" void kernel_launch(void* const* d_in, const int* in_sizes, int n_in,
                              void* d_out, int out_size, void* d_ws,
                              size_t ws_size, hipStream_t stream) {
  (void)in_sizes; (void)n_in; (void)out_size; (void)ws_size;
  const float* X = (const float*)d_in[0];
  const float* E_X = (const float*)d_in[1];
  const int* src = (const int*)d_in[2];
  const int* dst = (const int*)d_in[3];
  const float* snorm_n = (const float*)d_in[4];
  const float* snorm_e = (const float*)d_in[5];
  const float* WA = (const float*)d_in[6];  const float* bA = (const float*)d_in[7];
  const float* WB = (const float*)d_in[8];  const float* bB = (const float*)d_in[9];
  const float* WC = (const float*)d_in[10]; const float* bC = (const float*)d_in[11];
  const float* WD = (const float*)d_in[12]; const float* bD = (const float*)d_in[13];
  const float* WE = (const float*)d_in[14]; const float* bE = (const float*)d_in[15];
  const float* gamma_h = (const float*)d_in[16];
  const float* beta_h = (const float*)d_in[17];
  const float* gamma_e = (const float*)d_in[18];
  const float* beta_e = (const float*)d_in[19];

  float* outH = (float*)d_out;
  float* outE = outH + (size_t)N_NODES * D;

  const size_t ND = (size_t)N_NODES * D;
  float* ws = (float*)d_ws;
  float* AX = ws;            // [N,D]
  float* BX = ws + ND;       // [N,D]
  float* DXm = ws + 2 * ND;  // [N,D]
  float* EXm = ws + 3 * ND;  // [N,D]
  float* num = ws + 4 * ND;  // [N,D] (zeroed)
  float* den = ws + 5 * ND;  // [N,D] (zeroed)
  float* stats = ws + 6 * ND;  // 512 sums (zeroed) + 512 ab
  float* ab = stats + 512;

  hipMemsetAsync(num, 0, 2 * ND * sizeof(float), stream);  // num+den contiguous
  hipMemsetAsync(stats, 0, 512 * sizeof(float), stream);

  const dim3 blk(256);
  const int nodeBlocks = (N_NODES / 16 + 7) / 8;  // 391
  gemm_node_kernel<<<nodeBlocks, blk, 0, stream>>>(X, WA, bA, AX, N_NODES);
  gemm_node_kernel<<<nodeBlocks, blk, 0, stream>>>(X, WB, bB, BX, N_NODES);
  gemm_node_kernel<<<nodeBlocks, blk, 0, stream>>>(X, WD, bD, DXm, N_NODES);
  gemm_node_kernel<<<nodeBlocks, blk, 0, stream>>>(X, WE, bE, EXm, N_NODES);

  const int edgeBlocks = (N_EDGES / 16 + 7) / 8;  // 3907
  gemm_edge_kernel<<<edgeBlocks, blk, 0, stream>>>(
      E_X, WC, bC, DXm, EXm, BX, src, dst, snorm_e, num, den, outE, stats + 256);

  node_combine_kernel<<<512, blk, 0, stream>>>(X, AX, num, den, snorm_n, outH,
                                               stats);

  bn_params_kernel<<<1, blk, 0, stream>>>(stats, gamma_h, beta_h, gamma_e,
                                          beta_e, ab);

  finalize_kernel<<<512, blk, 0, stream>>>(X, outH, ab, ab + D, N_NODES * (D / 4));
  finalize_kernel<<<2048, blk, 0, stream>>>(E_X, outE, ab + 2 * D, ab + 3 * D,
                                            N_EDGES * (D / 4));
}